// GraphViT_62294205661425
// MI455X (gfx1250) — compile-verified
//
#include <hip/hip_runtime.h>
#include <math.h>

// ---------------- problem constants ----------------
static const int N_NODES = 20000;
static const int N_EDGE  = 60000;
static const int NC      = 1024;   // clusters
static const int NK      = 20;     // nodes per cluster
static const int POS_START = -3;

typedef __attribute__((ext_vector_type(2))) float v2f;
typedef __attribute__((ext_vector_type(8))) float v8f;

// force global (addrspace 1) loads -- keeps the WMMA loop off the FLAT path,
// so it only ties up LOADcnt (not DScnt) and skips aperture checks.
typedef __attribute__((address_space(1))) const float  g_f32;
typedef __attribute__((address_space(1))) const v2f    g_v2f;
__device__ __forceinline__ float ld_g (const float* p){ return *(g_f32*)p; }
__device__ __forceinline__ v2f   ld_g2(const float* p){ return *(g_v2f*)p; }

// ============================================================
// device helpers
// ============================================================
__device__ inline float dev_sigmoid(float x){ return 1.0f/(1.0f+expf(-x)); }

__device__ inline void emb16(float p, float* dst){
#pragma unroll
  for (int f = 0; f < 8; ++f) {
    float fr = exp2f((float)(POS_START + f)) * 3.14159265358979323846f;
    dst[f]     = cosf(p * fr);
    dst[8 + f] = sinf(p * fr);
  }
}

// ============================================================
// WMMA f32 GEMM building blocks.
// One wave computes a 16x64 output tile (4x v_wmma_f32_16x16x4_f32 per K-chunk).
// A fragment: lane l<16 -> row bm+l, K = k,k+1 ; lane l>=16 -> row bm+l-16, K = k+2,k+3
// B fragment: vgpr0 = B[k + 2*half][col], vgpr1 = B[k+1 + 2*half][col]
// B addressing is strength-reduced: 4 running per-lane column pointers, +4*ldb/step.
// ============================================================
__device__ __forceinline__ void mac_seg(const float* __restrict__ Arow,
                                        const float* __restrict__ Bseg, long ldb,
                                        int len, int hoff, const int* col,
                                        v8f (&acc)[4])
{
  const float* ap = Arow + hoff;
  const float* bp[4];
#pragma unroll
  for (int t = 0; t < 4; ++t) bp[t] = Bseg + (long)hoff * ldb + col[t];
  const long step = 4 * ldb;
  for (int k = 0; k < len; k += 4) {
    v2f a = ld_g2(ap + k);
#pragma unroll
    for (int t = 0; t < 4; ++t) {
      v2f b;
      b.x = ld_g(bp[t]);
      b.y = ld_g(bp[t] + ldb);
      acc[t] = __builtin_amdgcn_wmma_f32_16x16x4_f32(false, a, false, b,
                                                     (short)0, acc[t], false, false);
      bp[t] += step;
    }
  }
}

__device__ __forceinline__ void mac_seg_nt(const float* __restrict__ Arow,
                                           const float* __restrict__ Bt, long ldb,
                                           int len, int hoff, const int* col,
                                           v8f (&acc)[4])
{
  const float* ap = Arow + hoff;
  const float* bp[4];
#pragma unroll
  for (int t = 0; t < 4; ++t) bp[t] = Bt + (long)col[t] * ldb + hoff;
  for (int k = 0; k < len; k += 4) {
    v2f a = ld_g2(ap + k);
#pragma unroll
    for (int t = 0; t < 4; ++t) {
      v2f b = ld_g2(bp[t] + k);
      acc[t] = __builtin_amdgcn_wmma_f32_16x16x4_f32(false, a, false, b,
                                                     (short)0, acc[t], false, false);
    }
  }
}

// MODE 0: A is (M,K) row-major, stride lda.
// MODE 1: A row j = [NF[snd_j](d1) | NF[rcv_j](d1) | EF[j](d2)]  (A = NF, lda = d1)
// TRANSB: 0 -> B (K,N) row-major; 1 -> B (N,K) row-major.
// C = act(alpha * A@B + bias) + Radd
template<int MODE, int TRANSB>
__global__ __launch_bounds__(32) void gemm16t(
    const float* __restrict__ A, long lda,
    const int*   __restrict__ edges2,
    const float* __restrict__ EF, int d1, int d2,
    const float* __restrict__ Bw, long ldb,
    const float* __restrict__ bias,
    const float* Radd, long ldr,
    float* __restrict__ Cout, long ldc,
    int M, int K, int Nn, float alpha, int act)
{
  int bm = blockIdx.x * 16, bn = blockIdx.y * 64;
  int lane = threadIdx.x;
  int half = lane >> 4;
  int l16  = lane & 15;
  int hoff = half * 2;

  v8f acc[4];
#pragma unroll
  for (int t = 0; t < 4; ++t) acc[t] = (v8f){0.f,0.f,0.f,0.f,0.f,0.f,0.f,0.f};

  int col[4];
#pragma unroll
  for (int t = 0; t < 4; ++t) { int c = bn + t*16 + l16; col[t] = (c < Nn) ? c : (Nn - 1); }

  int rowA  = bm + l16;
  int rowLd = (rowA < M) ? rowA : (M - 1);   // clamp: loads stay in-range, stores guarded

  if (MODE == 0) {
    if (TRANSB == 0) mac_seg   (A + (long)rowLd*lda, Bw, ldb, K, hoff, col, acc);
    else             mac_seg_nt(A + (long)rowLd*lda, Bw, ldb, K, hoff, col, acc);
  } else {
    int s0 = edges2[rowLd*2], r0 = edges2[rowLd*2 + 1];
    mac_seg(A  + (long)s0*lda,   Bw,                   ldb, d1, hoff, col, acc);
    mac_seg(A  + (long)r0*lda,   Bw + (long)d1*ldb,    ldb, d1, hoff, col, acc);
    mac_seg(EF + (long)rowLd*d2, Bw + (long)2*d1*ldb,  ldb, d2, hoff, col, acc);
  }

#pragma unroll
  for (int t = 0; t < 4; ++t) {
    int colB = bn + t*16 + l16;
    if (colB >= Nn) continue;
#pragma unroll
    for (int r = 0; r < 8; ++r) {
      int row = bm + r + half*8;
      if (row >= M) continue;
      float v = acc[t][r] * alpha;
      if (bias) v += bias[colB];
      if (act == 1)      v = fmaxf(v, 0.f);
      else if (act == 2) v = tanhf(v);
      if (Radd) v += Radd[(long)row*ldr + colB];
      Cout[(long)row*ldc + colB] = v;
    }
  }
}

// fallback VALU GEMM for tiny K (13, 3) and tiny N (4)
__global__ void gemm_naive(const float* __restrict__ A, long lda,
                           const float* __restrict__ Bw, long ldb,
                           const float* __restrict__ bias,
                           float* __restrict__ Cm, long ldc,
                           int M, int K, int Nn, int act)
{
  long i = (long)blockIdx.x * blockDim.x + threadIdx.x;
  if (i >= (long)M * Nn) return;
  int r = (int)(i / Nn), c = (int)(i % Nn);
  float acc = bias ? bias[c] : 0.f;
  for (int k = 0; k < K; ++k) acc += A[(long)r*lda + k] * Bw[(long)k*ldb + c];
  if (act == 1)      acc = fmaxf(acc, 0.f);
  else if (act == 2) acc = tanhf(acc);
  Cm[(long)r*ldc + c] = acc;
}

// ============================================================
// LayerNorm over rows:  Y = (Radd?) + LN(X)*s + b   (one wave/row)
// ============================================================
__global__ __launch_bounds__(32) void ln_rows(const float* __restrict__ X,
                                              const float* __restrict__ s,
                                              const float* __restrict__ b,
                                              const float* Radd,
                                              float* Y, int D, int rows)
{
  int row = blockIdx.x;
  if (row >= rows) return;
  const float* x = X + (long)row * D;
  float s1 = 0.f, s2 = 0.f;
  for (int i = threadIdx.x; i < D; i += 32) { float v = x[i]; s1 += v; s2 += v*v; }
  for (int m = 16; m > 0; m >>= 1) { s1 += __shfl_xor(s1, m, 32); s2 += __shfl_xor(s2, m, 32); }
  float mean = s1 / D;
  float var  = s2 / D - mean * mean;
  float inv  = rsqrtf(var + 1e-5f);
  for (int i = threadIdx.x; i < D; i += 32) {
    float v = (x[i] - mean) * inv * s[i] + b[i];
    if (Radd) v += Radd[(long)row*D + i];
    Y[(long)row*D + i] = v;
  }
}

// ============================================================
// small elementwise / gather / scatter kernels
// ============================================================
__global__ void k_zero(float* p, long n){
  long i = (long)blockIdx.x*blockDim.x + threadIdx.x;
  if (i < n) p[i] = 0.f;
}
__global__ void k_axpy(float* __restrict__ y, const float* __restrict__ x, long n){
  long i = (long)blockIdx.x*blockDim.x + threadIdx.x;
  if (i < n) y[i] += x[i];
}
__global__ void k_embed2(const float* __restrict__ pos, int rows,
                         float* __restrict__ out, int ldo, int ooff){
  int n = blockIdx.x*blockDim.x + threadIdx.x;
  if (n >= rows) return;
  float t[16];
  emb16(pos[n*2+0], t);
#pragma unroll
  for (int i = 0; i < 16; ++i) out[(long)n*ldo + ooff + i] = t[i];
  emb16(pos[n*2+1], t);
#pragma unroll
  for (int i = 0; i < 16; ++i) out[(long)n*ldo + ooff + 16 + i] = t[i];
}
__global__ void k_centers(const float* __restrict__ mesh, const int* __restrict__ clus,
                          const float* __restrict__ mask, float* __restrict__ centers,
                          int C_, int K_){
  int c = blockIdx.x*blockDim.x + threadIdx.x;
  if (c >= C_) return;
  float sx = 0.f, sy = 0.f, sm = 0.f;
  for (int k = 0; k < K_; ++k) {
    int id = clus[c*K_ + k];
    sx += mesh[id*2]; sy += mesh[id*2+1];
    sm += mask[c*K_ + k];
  }
  float d = sm + 1e-8f;
  centers[c*2] = sx/d; centers[c*2+1] = sy/d;
}
__global__ void k_pe(const float* __restrict__ mesh, const int* __restrict__ clus,
                     const float* __restrict__ centers, float* __restrict__ pe,
                     int C_, int K_){
  int j = blockIdx.x*blockDim.x + threadIdx.x;
  if (j >= C_*K_) return;
  int c = j / K_;
  int id = clus[j];
  float dx = centers[c*2]   - mesh[id*2];
  float dy = centers[c*2+1] - mesh[id*2+1];
  float t[16];
  emb16(dx, t);
#pragma unroll
  for (int i = 0; i < 16; ++i) pe[(long)j*32 + i] = t[i];
  emb16(dy, t);
#pragma unroll
  for (int i = 0; i < 16; ++i) pe[(long)j*32 + 16 + i] = t[i];
}
__global__ void k_init_winner(int* w, int n){
  int i = blockIdx.x*blockDim.x + threadIdx.x;
  if (i < n) w[i] = -1;
}
__global__ void k_winner(const int* __restrict__ idx, int* __restrict__ w, int nj, int nmax){
  int j = blockIdx.x*blockDim.x + threadIdx.x;
  if (j >= nj) return;
  int t = idx[j];
  if (t < nmax) atomicMax(&w[t], j);
}
__global__ void k_rel(const float* __restrict__ pe, const int* __restrict__ winner,
                      float* __restrict__ mposenc, int n){
  long i = (long)blockIdx.x*blockDim.x + threadIdx.x;
  if (i >= (long)n*32) return;
  int nn = (int)(i / 32), c = (int)(i % 32);
  int j = winner[nn];
  int src = (j >= 0) ? j : nn;
  mposenc[(long)nn*64 + 32 + c] = pe[(long)src*32 + c];
}
__global__ void k_v0(const float* __restrict__ st, const float* __restrict__ nt,
                     float* __restrict__ V0, int n){
  long i = (long)blockIdx.x*blockDim.x + threadIdx.x;
  if (i >= (long)n*13) return;
  int nn = (int)(i/13), c = (int)(i%13);
  V0[i] = (c < 4) ? st[(long)nn*4 + c] : nt[(long)nn*9 + (c-4)];
}
__global__ void k_e0(const float* __restrict__ mesh, const int* __restrict__ edges,
                     float* __restrict__ E0, int ne){
  int e = blockIdx.x*blockDim.x + threadIdx.x;
  if (e >= ne) return;
  int s = edges[e*2], r = edges[e*2+1];
  float dx = mesh[s*2]   - mesh[r*2];
  float dy = mesh[s*2+1] - mesh[r*2+1];
  E0[e*3]   = dx;
  E0[e*3+1] = dy;
  E0[e*3+2] = sqrtf(dx*dx + dy*dy);
}
__global__ void k_concat2(const float* __restrict__ A, int d1,
                          const float* __restrict__ Bm, int d2,
                          float* __restrict__ out, int rows){
  long i = (long)blockIdx.x*blockDim.x + threadIdx.x;
  long D = d1 + d2;
  if (i >= (long)rows*D) return;
  int r = (int)(i / D), c = (int)(i % D);
  out[i] = (c < d1) ? A[(long)r*d1 + c] : Bm[(long)r*d2 + (c - d1)];
}
__global__ void k_concat3(const float* __restrict__ A, int d1,
                          const float* __restrict__ Bm, int d2,
                          const float* __restrict__ Cm, int d3,
                          float* __restrict__ out, int rows){
  long i = (long)blockIdx.x*blockDim.x + threadIdx.x;
  long D = d1 + d2 + d3;
  if (i >= (long)rows*D) return;
  int r = (int)(i / D), c = (int)(i % D);
  if (c < d1)            out[i] = A[(long)r*d1 + c];
  else if (c < d1 + d2)  out[i] = Bm[(long)r*d2 + (c - d1)];
  else                   out[i] = Cm[(long)r*d3 + (c - d1 - d2)];
}
__global__ void k_scatter_add(const int* __restrict__ edges,
                              const float* __restrict__ src,
                              float* __restrict__ agg, int ne){
  long i = (long)blockIdx.x*blockDim.x + threadIdx.x;
  if (i >= (long)ne*128) return;
  int e = (int)(i / 128), c = (int)(i % 128);
  int r = edges[e*2+1];
  atomicAdd(&agg[(long)r*128 + c], src[i]);
}
__global__ void k_feats(const float* __restrict__ V, const float* __restrict__ mposenc,
                        const int* __restrict__ clus, float* __restrict__ feats, int nj){
  long i = (long)blockIdx.x*blockDim.x + threadIdx.x;
  if (i >= (long)nj*192) return;
  int j = (int)(i / 192), c = (int)(i % 192);
  int node = clus[j];
  feats[i] = (c < 128) ? V[(long)node*128 + c] : mposenc[(long)node*64 + (c - 128)];
}
__global__ void k_ind_empty(const float* __restrict__ mask, int* __restrict__ ind,
                            int* __restrict__ empty, int C_, int K_){
  int c = blockIdx.x*blockDim.x + threadIdx.x;
  if (c >= C_) return;
  float s = 0.f;
  for (int k = 0; k < K_; ++k) s += mask[c*K_ + k];
  int id = (int)s - 1;
  if (id == -1) id = K_ - 1;
  ind[c] = id;
  empty[c] = (s == 0.f) ? 1 : 0;
}
__global__ void k_gru_gate(const float* __restrict__ gi, const float* __restrict__ gh,
                           float* __restrict__ h, float* __restrict__ wsel,
                           const int* __restrict__ ind, int t, int C_, int dh){
  long i = (long)blockIdx.x*blockDim.x + threadIdx.x;
  if (i >= (long)C_*dh) return;
  int c = (int)(i / dh), u = (int)(i % dh);
  long base = (long)c * 3 * dh;
  float ir = gi[base + u],        hr = gh[base + u];
  float iz = gi[base + dh + u],   hz = gh[base + dh + u];
  float in = gi[base + 2*dh + u], hn = gh[base + 2*dh + u];
  float r = dev_sigmoid(ir + hr);
  float z = dev_sigmoid(iz + hz);
  float n = tanhf(in + r * hn);
  float hv = (1.f - z) * n + z * h[i];
  h[i] = hv;
  if (ind[c] == t) wsel[i] = hv;
}
__global__ __launch_bounds__(32) void k_softmax(float* __restrict__ sc,
                                                const int* __restrict__ empty, int C_){
  int row = blockIdx.x;         // h*C + i
  int i = row % C_;
  float* p = sc + (long)row * C_;
  float mx = -3.4e38f;
  for (int j = threadIdx.x; j < C_; j += 32) {
    float v = p[j];
    if (j != i && empty[j]) { v = -1e9f; p[j] = v; }
    mx = fmaxf(mx, v);
  }
  for (int m = 16; m > 0; m >>= 1) mx = fmaxf(mx, __shfl_xor(mx, m, 32));
  float sum = 0.f;
  for (int j = threadIdx.x; j < C_; j += 32) { float e = expf(p[j] - mx); p[j] = e; sum += e; }
  for (int m = 16; m > 0; m >>= 1) sum += __shfl_xor(sum, m, 32);
  float inv = 1.f / sum;
  for (int j = threadIdx.x; j < C_; j += 32) p[j] *= inv;
}
__global__ void k_scatter_ws(const int* __restrict__ winner, const float* __restrict__ W,
                             float* __restrict__ Ws, int n, int K_){
  long i = (long)blockIdx.x*blockDim.x + threadIdx.x;
  if (i >= (long)n*512) return;
  int nn = (int)(i / 512), c = (int)(i % 512);
  int j = winner[nn];
  int row = ((j >= 0) ? j : nn) / K_;
  Ws[i] = W[(long)row*512 + c];
}
__global__ void k_finalize(const float* __restrict__ state, const float* __restrict__ ntype,
                           const float* __restrict__ next_out, float* __restrict__ out, int n){
  long i = (long)blockIdx.x*blockDim.x + threadIdx.x;
  if (i >= (long)n*4) return;
  int nn = (int)(i / 4);
  const float* st0 = state;
  const float* st1 = state + (long)n*4;
  const float* nt1 = ntype + (long)n*9;
  float s0 = st0[i], s1 = st1[i], no = next_out[i];
  float ns = s0 + no;
  bool m = (nt1[(long)nn*9+4] == 1.0f) || (nt1[(long)nn*9+6] == 1.0f) || (nt1[(long)nn*9+2] == 1.0f);
  if (m) ns = s1;
  out[i]                 = s0;       // state_hat[t=1]
  out[(long)n*4   + i]   = ns;       // state_hat[t=2]
  out[(long)n*8   + i]   = no;       // output_hat
  out[(long)n*12  + i]   = s1 - s0;  // target
}

// ============================================================
// host
// ============================================================
extern "C" void kernel_launch(void* const* d_in, const int* in_sizes, int n_in,
                              void* d_out, int out_size, void* d_ws, size_t ws_size,
                              hipStream_t stream)
{
  const int N = N_NODES, NE = N_EDGE, C = NC, K = NK;
  const float* mesh  = (const float*)d_in[0];   // (1,2,N,2) -> t=0
  const float* state = (const float*)d_in[1];   // (1,2,N,4)
  const float* ntype = (const float*)d_in[2];   // (1,2,N,9)
  const float* cmask = (const float*)d_in[3];   // (1,2,C,K) -> t=0
  const float* P     = (const float*)d_in[4];   // flat params (tree-flatten order)
  const int*   edges = (const int*)d_in[5];     // (1,2,NE,2) -> t=0
  const int*   clus  = (const int*)d_in[6];     // (1,2,C,K)  -> t=0
  float* out = (float*)d_out;
  (void)in_sizes; (void)n_in; (void)out_size; (void)ws_size;

  // ---------- parameter offsets (jax tree-flatten: sorted dict keys) ----------
  size_t po = 0;
  auto take = [&](size_t n)->size_t { size_t o = po; po += n; return o; };
  size_t a_in_b[4], a_in_w[4], a_lin_b[4], a_lin_w[4], a_ln1_b[4], a_ln1_s[4],
         a_ln2_b[4], a_ln2_s[4], a_l0_b[4], a_l0_w[4], a_l1_b[4], a_l1_w[4],
         a_out_b[4], a_out_w[4];
  for (int i = 0; i < 4; ++i) {                       // 'attn' (list of blocks)
    a_in_b[i] = take(1632);  a_in_w[i] = take((size_t)544*1632);
    a_lin_b[i]= take(512);   a_lin_w[i]= take((size_t)544*512);
    a_ln1_b[i]= take(512);   a_ln1_s[i]= take(512);
    a_ln2_b[i]= take(512);   a_ln2_s[i]= take(512);
    a_l0_b[i] = take(512);   a_l0_w[i] = take((size_t)512*512);
    a_l1_b[i] = take(512);   a_l1_w[i] = take((size_t)512*512);
    a_out_b[i]= take(544);   a_out_w[i]= take((size_t)544*544);
  }
  // 'encoder': enc_edge, enc_node, gns
  size_t ee_l0_b = take(128), ee_l0_w = take(3*128), ee_l1_b = take(128), ee_l1_w = take(128*128);
  size_t en_l0_b = take(128), en_l0_w = take(13*128), en_l1_b = take(128), en_l1_w = take(128*128);
  size_t ge_l0_b[4], ge_l0_w[4], ge_l1_b[4], ge_l1_w[4], ge_ln_b[4], ge_ln_s[4];
  size_t gn_l0_b[4], gn_l0_w[4], gn_l1_b[4], gn_l1_w[4], gn_ln_b[4], gn_ln_s[4];
  for (int i = 0; i < 4; ++i) {
    ge_l0_b[i] = take(128); ge_l0_w[i] = take((size_t)512*128);
    ge_l1_b[i] = take(128); ge_l1_w[i] = take(128*128);
    ge_ln_b[i] = take(128); ge_ln_s[i] = take(128);
    gn_l0_b[i] = take(128); gn_l0_w[i] = take((size_t)320*128);
    gn_l1_b[i] = take(128); gn_l1_w[i] = take(128*128);
    gn_ln_b[i] = take(128); gn_ln_s[i] = take(128);
  }
  // 'ln'
  size_t fln_b = take(512), fln_s = take(512);
  // 'pool': gru (bhh,bih,whh,wih), linear_rnn
  size_t gru_bhh = take(1536), gru_bih = take(1536);
  size_t gru_whh = take((size_t)512*1536), gru_wih = take((size_t)192*1536);
  size_t lr_l0_b = take(128), lr_l0_w = take((size_t)512*128);
  size_t lr_l1_b = take(512), lr_l1_w = take((size_t)128*512);
  // 'retrieve': fm0, fm1, fm2, gnn
  size_t fm0_b = take(128), fm0_w = take(128*128);
  size_t fm1_b = take(128), fm1_w = take(128*128);
  size_t fm2_b = take(4),   fm2_w = take(128*4);
  size_t re_l0_b = take(128), re_l0_w = take((size_t)1536*128);
  size_t re_l1_b = take(128), re_l1_w = take(128*128);
  size_t re_ln_b = take(128), re_ln_s = take(128);
  size_t rn_l0_b = take(128), rn_l0_w = take((size_t)832*128);
  size_t rn_l1_b = take(128), rn_l1_w = take(128*128);
  size_t rn_ln_b = take(128), rn_ln_s = take(128);

  // ---------- workspace carve ----------
  char* wsb = (char*)d_ws;
  size_t wo = 0;
  auto alloc = [&](size_t nfloats)->float* {
    float* p = (float*)(wsb + wo);
    wo += ((nfloats * 4 + 255) / 256) * 256;
    return p;
  };
  float* mposenc = alloc((size_t)N*64);
  float* centers = alloc((size_t)C*2);
  float* cposenc = alloc((size_t)C*32);
  float* pe      = alloc((size_t)C*K*32);
  int*   winner  = (int*)alloc(N);
  float* V0      = alloc((size_t)N*13);
  float* E0      = alloc((size_t)NE*3);
  float* V       = alloc((size_t)N*128);
  float* Vh      = alloc((size_t)N*128);
  float* Vnew    = alloc((size_t)N*128);
  float* E       = alloc((size_t)NE*128);
  float* Ehid    = alloc((size_t)NE*128);
  float* Enew    = alloc((size_t)NE*128);
  float* Vin     = alloc((size_t)N*192);
  float* agg     = alloc((size_t)N*128);
  float* vf_in   = alloc((size_t)N*320);
  float* feats   = alloc((size_t)C*K*192);
  float* gi      = alloc((size_t)C*1536);
  float* gh      = alloc((size_t)C*1536);
  float* h       = alloc((size_t)C*512);
  float* wsel    = alloc((size_t)C*512);
  float* lrh     = alloc((size_t)C*128);
  int*   ind     = (int*)alloc(C);
  int*   empty   = (int*)alloc(C);
  float* W       = alloc((size_t)C*512);
  float* Wln     = alloc((size_t)C*512);
  float* xcat    = alloc((size_t)C*544);
  float* qkv     = alloc((size_t)C*1632);
  float* scores  = alloc((size_t)4*C*C);
  float* omat    = alloc((size_t)C*544);
  float* W2      = alloc((size_t)C*544);
  float* W3      = alloc((size_t)C*512);
  float* mlph    = alloc((size_t)C*512);
  float* Wfin    = alloc((size_t)C*512);
  float* Ws      = alloc((size_t)N*512);
  float* nodes   = alloc((size_t)N*704);
  float* nf_in   = alloc((size_t)N*832);
  float* vr      = alloc((size_t)N*128);
  float* next_o  = alloc((size_t)N*4);

  auto g1 = [&](long n)->dim3 { return dim3((unsigned)((n + 255) / 256)); };
  auto gemm = [&](const float* A, long lda, const float* Bw, long ldb,
                  const float* bias, const float* Radd, long ldr,
                  float* Cm, long ldc, int M, int Kd, int Nn, float alpha, int act){
    dim3 g((M + 15) / 16, (Nn + 63) / 64);
    gemm16t<0,0><<<g, 32, 0, stream>>>(A, lda, nullptr, nullptr, 0, 0,
                                       Bw, ldb, bias, Radd, ldr,
                                       Cm, ldc, M, Kd, Nn, alpha, act);
  };
  auto gemmNT = [&](const float* A, long lda, const float* Bt, long ldb,
                    float* Cm, long ldc, int M, int Kd, int Nn, float alpha){
    dim3 g((M + 15) / 16, (Nn + 63) / 64);
    gemm16t<0,1><<<g, 32, 0, stream>>>(A, lda, nullptr, nullptr, 0, 0,
                                       Bt, ldb, nullptr, nullptr, 0,
                                       Cm, ldc, M, Kd, Nn, alpha, 0);
  };
  auto gemmEdge = [&](const int* eg, const float* NF, int d1, const float* EF, int d2,
                      const float* Bw, const float* bias, float* Cm,
                      int M, int Kd, int Nn, int act){
    dim3 g((M + 15) / 16, (Nn + 63) / 64);
    gemm16t<1,0><<<g, 32, 0, stream>>>(NF, d1, eg, EF, d1, d2,
                                       Bw, (long)Nn, bias, nullptr, 0,
                                       Cm, (long)Nn, M, Kd, Nn, 1.0f, act);
  };

  // ================= stage 1: positional encoder =================
  k_embed2<<<g1(N), 256, 0, stream>>>(mesh, N, mposenc, 64, 0);
  k_centers<<<g1(C), 256, 0, stream>>>(mesh, clus, cmask, centers, C, K);
  k_embed2<<<g1(C), 256, 0, stream>>>(centers, C, cposenc, 32, 0);
  k_pe<<<g1((long)C*K), 256, 0, stream>>>(mesh, clus, centers, pe, C, K);
  k_init_winner<<<g1(N), 256, 0, stream>>>(winner, N);
  k_winner<<<g1((long)C*K), 256, 0, stream>>>(clus, winner, C*K, N);
  k_rel<<<g1((long)N*32), 256, 0, stream>>>(pe, winner, mposenc, N);

  // ================= stage 2: encoder =================
  k_v0<<<g1((long)N*13), 256, 0, stream>>>(state, ntype, V0, N);
  gemm_naive<<<g1((long)N*128), 256, 0, stream>>>(V0, 13, P+en_l0_w, 128, P+en_l0_b, Vh, 128, N, 13, 128, 1);
  gemm(Vh, 128, P+en_l1_w, 128, P+en_l1_b, nullptr, 0, V, 128, N, 128, 128, 1.f, 0);
  k_e0<<<g1(NE), 256, 0, stream>>>(mesh, edges, E0, NE);
  gemm_naive<<<g1((long)NE*128), 256, 0, stream>>>(E0, 3, P+ee_l0_w, 128, P+ee_l0_b, Ehid, 128, NE, 3, 128, 1);
  gemm(Ehid, 128, P+ee_l1_w, 128, P+ee_l1_b, nullptr, 0, E, 128, NE, 128, 128, 1.f, 0);

  for (int g = 0; g < 4; ++g) {
    k_concat2<<<g1((long)N*192), 256, 0, stream>>>(V, 128, mposenc, 64, Vin, N);
    gemmEdge(edges, Vin, 192, E, 128, P+ge_l0_w[g], P+ge_l0_b[g], Ehid, NE, 512, 128, 1);
    gemm(Ehid, 128, P+ge_l1_w[g], 128, P+ge_l1_b[g], nullptr, 0, Enew, 128, NE, 128, 128, 1.f, 0);
    ln_rows<<<NE, 32, 0, stream>>>(Enew, P+ge_ln_s[g], P+ge_ln_b[g], nullptr, Ehid, 128, NE); // Ehid = LN(e_new)
    k_zero<<<g1((long)N*128), 256, 0, stream>>>(agg, (long)N*128);
    k_scatter_add<<<g1((long)NE*128), 256, 0, stream>>>(edges, Ehid, agg, NE);
    k_concat2<<<g1((long)N*320), 256, 0, stream>>>(Vin, 192, agg, 128, vf_in, N);
    gemm(vf_in, 320, P+gn_l0_w[g], 128, P+gn_l0_b[g], nullptr, 0, Vh, 128, N, 320, 128, 1.f, 1);
    gemm(Vh, 128, P+gn_l1_w[g], 128, P+gn_l1_b[g], nullptr, 0, Vnew, 128, N, 128, 128, 1.f, 0);
    ln_rows<<<N, 32, 0, stream>>>(Vnew, P+gn_ln_s[g], P+gn_ln_b[g], V, V, 128, N); // V += LN(v_new)
    k_axpy<<<g1((long)NE*128), 256, 0, stream>>>(E, Ehid, (long)NE*128);           // E += LN(e_new)
  }

  // ================= stage 3: graph pooling (GRU) =================
  k_feats<<<g1((long)C*K*192), 256, 0, stream>>>(V, mposenc, clus, feats, C*K);
  k_ind_empty<<<g1(C), 256, 0, stream>>>(cmask, ind, empty, C, K);
  k_zero<<<g1((long)C*512), 256, 0, stream>>>(h, (long)C*512);
  for (int t = 0; t < K; ++t) {
    gemm(feats + (long)t*192, (long)K*192, P+gru_wih, 1536, P+gru_bih, nullptr, 0, gi, 1536, C, 192, 1536, 1.f, 0);
    gemm(h, 512, P+gru_whh, 1536, P+gru_bhh, nullptr, 0, gh, 1536, C, 512, 1536, 1.f, 0);
    k_gru_gate<<<g1((long)C*512), 256, 0, stream>>>(gi, gh, h, wsel, ind, t, C, 512);
  }
  gemm(wsel, 512, P+lr_l0_w, 128, P+lr_l0_b, nullptr, 0, lrh, 128, C, 512, 128, 1.f, 1);
  gemm(lrh, 128, P+lr_l1_w, 512, P+lr_l1_b, nullptr, 0, W, 512, C, 128, 512, 1.f, 0);

  // ================= stage 4: attention blocks =================
  const float att_alpha = 1.0f / sqrtf(136.0f);
  for (int i = 0; i < 4; ++i) {
    ln_rows<<<C, 32, 0, stream>>>(W, P+a_ln1_s[i], P+a_ln1_b[i], nullptr, Wln, 512, C);
    k_concat2<<<g1((long)C*544), 256, 0, stream>>>(Wln, 512, cposenc, 32, xcat, C);
    gemm(xcat, 544, P+a_in_w[i], 1632, P+a_in_b[i], nullptr, 0, qkv, 1632, C, 544, 1632, 1.f, 0);
    for (int hh = 0; hh < 4; ++hh)
      gemmNT(qkv + hh*136, 1632, qkv + 544 + hh*136, 1632,
             scores + (long)hh*C*C, C, C, 136, C, att_alpha);
    k_softmax<<<4*C, 32, 0, stream>>>(scores, empty, C);
    for (int hh = 0; hh < 4; ++hh)
      gemm(scores + (long)hh*C*C, C, qkv + 1088 + hh*136, 1632, nullptr, nullptr, 0,
           omat + hh*136, 544, C, C, 136, 1.f, 0);
    gemm(omat, 544, P+a_out_w[i], 544, P+a_out_b[i], nullptr, 0, W2, 544, C, 544, 544, 1.f, 0);
    gemm(W2, 544, P+a_lin_w[i], 512, P+a_lin_b[i], W, 512, W3, 512, C, 544, 512, 1.f, 0);
    ln_rows<<<C, 32, 0, stream>>>(W3, P+a_ln2_s[i], P+a_ln2_b[i], nullptr, Wln, 512, C);
    gemm(Wln, 512, P+a_l0_w[i], 512, P+a_l0_b[i], nullptr, 0, mlph, 512, C, 512, 512, 1.f, 1);
    gemm(mlph, 512, P+a_l1_w[i], 512, P+a_l1_b[i], W3, 512, W, 512, C, 512, 512, 1.f, 0);
  }
  ln_rows<<<C, 32, 0, stream>>>(W, P+fln_s, P+fln_b, nullptr, Wfin, 512, C);

  // ================= stage 5: graph retrieve =================
  k_scatter_ws<<<g1((long)N*512), 256, 0, stream>>>(winner, Wfin, Ws, N, K);
  k_concat3<<<g1((long)N*704), 256, 0, stream>>>(V, 128, Ws, 512, mposenc, 64, nodes, N);
  gemmEdge(edges, nodes, 704, E, 128, P+re_l0_w, P+re_l0_b, Ehid, NE, 1536, 128, 1);
  gemm(Ehid, 128, P+re_l1_w, 128, P+re_l1_b, nullptr, 0, Enew, 128, NE, 128, 128, 1.f, 0);
  ln_rows<<<NE, 32, 0, stream>>>(Enew, P+re_ln_s, P+re_ln_b, nullptr, Ehid, 128, NE);
  k_zero<<<g1((long)N*128), 256, 0, stream>>>(agg, (long)N*128);
  k_scatter_add<<<g1((long)NE*128), 256, 0, stream>>>(edges, Ehid, agg, NE);
  k_concat2<<<g1((long)N*832), 256, 0, stream>>>(nodes, 704, agg, 128, nf_in, N);
  gemm(nf_in, 832, P+rn_l0_w, 128, P+rn_l0_b, nullptr, 0, Vh, 128, N, 832, 128, 1.f, 1);
  gemm(Vh, 128, P+rn_l1_w, 128, P+rn_l1_b, nullptr, 0, Vnew, 128, N, 128, 128, 1.f, 0);
  ln_rows<<<N, 32, 0, stream>>>(Vnew, P+rn_ln_s, P+rn_ln_b, nullptr, vr, 128, N);
  gemm(vr, 128, P+fm0_w, 128, P+fm0_b, nullptr, 0, Vh, 128, N, 128, 128, 1.f, 2);
  gemm(Vh, 128, P+fm1_w, 128, P+fm1_b, nullptr, 0, Vnew, 128, N, 128, 128, 1.f, 2);
  gemm_naive<<<g1((long)N*4), 256, 0, stream>>>(Vnew, 128, P+fm2_w, 4, P+fm2_b, next_o, 4, N, 128, 4, 0);

  // ================= stage 6: finalize outputs =================
  k_finalize<<<g1((long)N*4), 256, 0, stream>>>(state, ntype, next_o, out, N);
}